// SPDTangentSpace_41824391528794
// MI455X (gfx1250) — compile-verified
//
#include <hip/hip_runtime.h>

// SPD matrix logarithm via inverse scaling-and-squaring:
//   log(A) = 2^k * log(A^(1/2^k)),  sqrt by coupled Newton-Schulz (matmul-only),
//   log(I+E) by degree-9 Horner series.  All matmuls use V_WMMA_F32_16X16X4_F32.
// One workgroup (16 wave32s) per matrix; 4x 64KB fp32 matrices in dynamic LDS.
// All LDS traffic goes through the file-scope addrspace(3) symbol `smem`
// indexed by integer offsets, so it lowers to ds_load/ds_store (not flat_*).

typedef __attribute__((ext_vector_type(2))) float v2f;
typedef __attribute__((ext_vector_type(8))) float v8f;

#define NN 128
#define ME (NN * NN)          // 16384 elements = 64 KB
#define NTHREADS 512
#define NWAVES 16
#define KSQRT 6               // number of repeated square roots
#define NSITER 16             // Newton-Schulz iterations per sqrt

extern __shared__ float smem[];          // 4 * ME floats (256 KB)
__shared__ float red[NWAVES];            // reduction scratch (static LDS)

// smem[cO..] = alpha*(smem[aO..] @ smem[bO..]) + beta*I, 128x128 fp32.
// 16 waves over the 8x8 grid of 16x16 tiles; each wave owns a 2x2 tile block
// (rows (w>>2)*2.., cols (w&3)*2..) so both A and B fragments are reused 2x:
// 4 ds loads feed 4 wmma per k-step.
__device__ __forceinline__ void mm128(int cO, int aO, int bO,
                                      float alpha, float beta,
                                      int wave, int lane)
{
    __syncthreads();                       // inputs fully written
    const int half = lane >> 4;            // 0/1
    const int m    = lane & 15;
    const int row0 = (wave >> 2) * 32;     // first row of 2x2 tile block
    const int col0 = (wave & 3) * 32;      // first col of 2x2 tile block

    v8f acc[4] = {};                       // [rowTile*2 + colTile]

    const int arow0 = aO + (row0 + m) * NN + 2 * half;        // + k
    const int arow1 = arow0 + 16 * NN;
    #pragma unroll 4
    for (int k = 0; k < NN; k += 4) {
        // A fragments 16x4 (two row tiles): contiguous pair -> ds_load_b64
        v2f a0, a1;
        a0.x = smem[arow0 + k];
        a0.y = smem[arow0 + k + 1];
        a1.x = smem[arow1 + k];
        a1.y = smem[arow1 + k + 1];
        // B fragments 4x16 (two col tiles)
        const int b0 = bO + (k + 2 * half) * NN + m;
        v2f bf0, bf1;
        bf0.x = smem[b0 + col0];
        bf0.y = smem[b0 + NN + col0];
        bf1.x = smem[b0 + col0 + 16];
        bf1.y = smem[b0 + NN + col0 + 16];

        acc[0] = __builtin_amdgcn_wmma_f32_16x16x4_f32(
            false, a0, false, bf0, (short)0, acc[0], false, false);
        acc[1] = __builtin_amdgcn_wmma_f32_16x16x4_f32(
            false, a0, false, bf1, (short)0, acc[1], false, false);
        acc[2] = __builtin_amdgcn_wmma_f32_16x16x4_f32(
            false, a1, false, bf0, (short)0, acc[2], false, false);
        acc[3] = __builtin_amdgcn_wmma_f32_16x16x4_f32(
            false, a1, false, bf1, (short)0, acc[3], false, false);
    }

    // C/D layout: VGPR j -> row = tileRow + 8*half + j, col = tileCol + m
    #pragma unroll
    for (int i = 0; i < 2; ++i) {
        #pragma unroll
        for (int jt = 0; jt < 2; ++jt) {
            const v8f& a = acc[i * 2 + jt];
            const int col = col0 + jt * 16 + m;
            #pragma unroll
            for (int j = 0; j < 8; ++j) {
                const int row = row0 + i * 16 + half * 8 + j;
                float v = alpha * a[j];
                if (row == col) v += beta;
                smem[cO + row * NN + col] = v;
            }
        }
    }
    __syncthreads();                       // C visible before next consumer
}

__device__ __forceinline__ void scale_copy(int dO, int sO, float s, int tid)
{
    for (int i = tid; i < ME; i += NTHREADS) smem[dO + i] = s * smem[sO + i];
    __syncthreads();
}

__device__ __forceinline__ void scaled_plus_diag(int dO, int sO,
                                                 float s, float d, int tid)
{
    for (int i = tid; i < ME; i += NTHREADS) {
        const int r = i >> 7, c = i & (NN - 1);
        smem[dO + i] = s * smem[sO + i] + ((r == c) ? d : 0.0f);
    }
    __syncthreads();
}

__device__ __forceinline__ void set_identity(int dO, int tid)
{
    for (int i = tid; i < ME; i += NTHREADS) {
        const int r = i >> 7, c = i & (NN - 1);
        smem[dO + i] = (r == c) ? 1.0f : 0.0f;
    }
    __syncthreads();
}

__device__ __forceinline__ float frob(int mO, int tid)
{
    float s = 0.0f;
    for (int i = tid; i < ME; i += NTHREADS) {
        const float v = smem[mO + i];
        s = fmaf(v, v, s);
    }
    #pragma unroll
    for (int off = 16; off; off >>= 1) s += __shfl_xor(s, off, 32);
    if ((tid & 31) == 0) red[tid >> 5] = s;
    __syncthreads();
    if (tid == 0) {
        float t = 0.0f;
        for (int w = 0; w < NWAVES; ++w) t += red[w];
        red[0] = t;
    }
    __syncthreads();
    const float r = red[0];
    __syncthreads();
    return sqrtf(r);
}

__global__ __launch_bounds__(NTHREADS, 1)
void spd_logm_wmma(const float* __restrict__ X, float* __restrict__ out)
{
    const int tid  = threadIdx.x;
    const int wave = tid >> 5;
    const int lane = tid & 31;

    // ---- load matrix: global_load_b128 -> ds_store_b128 ----
    {
        const float4* src = reinterpret_cast<const float4*>(X + (size_t)blockIdx.x * ME);
        for (int i = tid; i < ME / 4; i += NTHREADS) {
            const float4 v = src[i];
            smem[4 * i + 0] = v.x;
            smem[4 * i + 1] = v.y;
            smem[4 * i + 2] = v.z;
            smem[4 * i + 3] = v.w;
        }
        __syncthreads();
    }

    int iM = 0;                            // buffer index 0..3; offset = idx*ME

    // ---- k repeated square roots: M <- sqrt(M) via coupled Newton-Schulz ----
    for (int s = 0; s < KSQRT; ++s) {
        const float c  = frob(iM * ME, tid);        // spectrum of M/c in (0,1]
        const float sc = sqrtf(c);
        int iY = (iM + 1) & 3, iZ = (iM + 2) & 3, iT = (iM + 3) & 3;
        scale_copy(iY * ME, iM * ME, 1.0f / c, tid);  // Y0 = M/c
        set_identity(iZ * ME, tid);                   // Z0 = I
        int f0 = iM, f1 = iT;                         // free buffers
        for (int it = 0; it < NSITER; ++it) {
            mm128(f1 * ME, iZ * ME, iY * ME, -1.0f, 3.0f, wave, lane); // T = 3I - Z*Y
            mm128(f0 * ME, iY * ME, f1 * ME,  0.5f, 0.0f, wave, lane); // Ynew = .5*Y*T
            mm128(iY * ME, f1 * ME, iZ * ME,  0.5f, 0.0f, wave, lane); // Znew = .5*T*Z
            const int oy = iY;                        // rotate buffer roles
            iY = f0;
            f0 = iZ;
            iZ = oy;
        }
        scale_copy(f0 * ME, iY * ME, sc, tid);        // M = sqrt(c) * Y
        iM = f0;
    }

    // ---- log(M) = log(I + E), E = M - I, degree-9 Horner (c_n = (-1)^(n+1)/n)
    int iE = (iM + 1) & 3, iS = (iM + 2) & 3, iT2 = (iM + 3) & 3;
    scaled_plus_diag(iE * ME, iM * ME, 1.0f, -1.0f, tid);               // E = M - I
    scaled_plus_diag(iS * ME, iE * ME, 1.0f / 9.0f, -1.0f / 8.0f, tid); // S = c9*E + c8*I
    for (int n = 7; n >= 1; --n) {
        const float cn = ((n & 1) ? 1.0f : -1.0f) / (float)n;
        mm128(iT2 * ME, iS * ME, iE * ME, 1.0f, cn, wave, lane);        // S = S*E + cn*I
        const int t = iS; iS = iT2; iT2 = t;
    }
    const float two_k = (float)(1 << KSQRT);
    mm128(iM * ME, iS * ME, iE * ME, two_k, 0.0f, wave, lane);          // L = 2^k * S * E

    // ---- half-vectorize: upper triangle (incl. diagonal), row-major ----
    const int lO = iM * ME;
    float* ob = out + (size_t)blockIdx.x * (NN * (NN + 1) / 2);
    for (int r = wave; r < NN; r += NWAVES) {
        const int off = r * NN - (r * (r - 1)) / 2;
        for (int c = r + lane; c < NN; c += 32) {
            ob[off + (c - r)] = smem[lO + r * NN + c];
        }
    }
}

extern "C" void kernel_launch(void* const* d_in, const int* in_sizes, int n_in,
                              void* d_out, int out_size, void* d_ws, size_t ws_size,
                              hipStream_t stream)
{
    const float* x = (const float*)d_in[0];
    float* out = (float*)d_out;
    const int batch = in_sizes[0] / ME;                 // 2048
    const size_t shmem = (size_t)(4 * ME) * sizeof(float);   // 256 KB dynamic (<320 KB WGP LDS)
    (void)hipFuncSetAttribute(reinterpret_cast<const void*>(spd_logm_wmma),
                              hipFuncAttributeMaxDynamicSharedMemorySize, (int)shmem);
    spd_logm_wmma<<<batch, NTHREADS, shmem, stream>>>(x, out);
}